// LegendreBasis_x_57526791963012
// MI455X (gfx1250) — compile-verified
//
#include <hip/hip_runtime.h>
#include <stdint.h>

// Packed fp32 vectors: target CDNA5 V_PK_FMA_F32 / V_PK_MUL_F32 (VOP3P).
typedef float v2f __attribute__((ext_vector_type(2)));
typedef float v4f __attribute__((ext_vector_type(4)));

#if defined(__has_builtin)
#if __has_builtin(__builtin_elementwise_fma)
#define V2FMA(a, b, c) __builtin_elementwise_fma((a), (b), (c))
#endif
#endif
#ifndef V2FMA
static __device__ __forceinline__ v2f v2fma_fallback(v2f a, v2f b, v2f c) {
  v2f r;
  r.x = __builtin_fmaf(a.x, b.x, c.x);
  r.y = __builtin_fmaf(a.y, b.y, c.y);
  return r;
}
#define V2FMA(a, b, c) v2fma_fallback((a), (b), (c))
#endif

#if defined(__has_builtin)
#if __has_builtin(__builtin_amdgcn_global_load_async_to_lds_b128)
#define HAVE_ASYNC_LDS 1
#endif
#endif

// Exact parameter type from the hipcc diagnostic:
//   __attribute__((__vector_size__(4 * sizeof(int)))) int __device__ *
typedef int v4i_vs __attribute__((__vector_size__(4 * sizeof(int))));
typedef __attribute__((address_space(1))) v4i_vs *gv4p;   // global (AS1)
typedef __attribute__((address_space(3))) v4i_vs *lv4p;   // LDS (AS3)

// Constant-address-space view for the uniform fallback path.
typedef const float __attribute__((address_space(4))) *cfp4;

#define DEG128 128
#define CHUNK 8

static __device__ __forceinline__ int imin(int a, int b) {
  return a < b ? a : b;
}

// ---------------------------------------------------------------------------
// bias(x) = sum_{k<128} w_k P_k(x);  P_{k+1} = a_k x P_k - b_k P_{k-1},
//   a_k = (2k+1)/(k+1), b_k = k/(k+1).
//
// wtab (prebuilt in global scratch by build_wtab) holds, per k, 32 B:
//   { w,w, a,a | nb,nb, pad,pad }
// The block copies the 4 KB table into LDS via the gfx1250 async global->LDS
// DMA path (ASYNCcnt-tracked), then the packed math consumes it via ds_load
// into VGPR pairs: the VALU stream is pure v_pk_* (no SGPR-literal pressure,
// no spills) and the DS traffic rides the separate LDS pipe
// (uniform-address broadcast, conflict-free).
//
// 4 elements per thread as two packed-f32 chains (ILP for the serial
// recursion chain). Loads are index-clamped; stores guarded at the end.
// ---------------------------------------------------------------------------
__global__ __launch_bounds__(256) void legendre128_tab(
    const float *__restrict__ pos, const v4f *__restrict__ wtab,
    float *__restrict__ out, int n) {
  __shared__ v4f tab4[2 * DEG128];  // 4 KB, one 16B chunk per thread

#if defined(HAVE_ASYNC_LDS)
  {
    gv4p g = (gv4p)((uintptr_t)wtab + 16u * threadIdx.x);
    lv4p l = (lv4p)(v4i_vs *)&tab4[threadIdx.x];
    __builtin_amdgcn_global_load_async_to_lds_b128(g, l, 0, 0);
    asm volatile("s_wait_asynccnt 0x0" ::: "memory");
  }
#else
  tab4[threadIdx.x] = wtab[threadIdx.x];
#endif
  __syncthreads();

  const v2f *tab = (const v2f *)tab4;

  const int i0 = (blockIdx.x * blockDim.x + threadIdx.x) * 4;
  if (i0 >= n) return;

  // Clamped gather of 4 x-values (stride 3); duplicated tail lanes are
  // simply not stored.
  const int last = n - 1;
  v2f xA, xB;
  xA.x = pos[3 * i0];
  xA.y = pos[3 * imin(i0 + 1, last)];
  xB.x = pos[3 * imin(i0 + 2, last)];
  xB.y = pos[3 * imin(i0 + 3, last)];

  v2f PmA = {0.0f, 0.0f}, PnA = {1.0f, 1.0f}, accA = {0.0f, 0.0f};
  v2f PmB = {0.0f, 0.0f}, PnB = {1.0f, 1.0f}, accB = {0.0f, 0.0f};

#pragma unroll 1
  for (int c = 0; c < DEG128; c += CHUNK) {
#pragma unroll
    for (int j = 0; j < CHUNK; ++j) {
      const int k = c + j;
      const v2f wv = tab[4 * k + 0];  // ds_load (imm offset within chunk)
      const v2f av = tab[4 * k + 1];
      const v2f nbv = tab[4 * k + 2];

      accA = V2FMA(wv, PnA, accA);  // v_pk_fma_f32
      accB = V2FMA(wv, PnB, accB);  // v_pk_fma_f32
      const v2f tA = xA * PnA;      // v_pk_mul_f32
      const v2f tB = xB * PnB;      // v_pk_mul_f32
      const v2f uA = nbv * PmA;     // v_pk_mul_f32
      const v2f uB = nbv * PmB;     // v_pk_mul_f32
      PmA = PnA;
      PmB = PnB;
      PnA = V2FMA(av, tA, uA);      // v_pk_fma_f32
      PnB = V2FMA(av, tB, uB);      // v_pk_fma_f32
    }
  }

  if (i0 + 3 < n) {
    v4f r = {accA.x, accA.y, accB.x, accB.y};
    *reinterpret_cast<v4f *>(out + i0) = r;  // 16B-aligned b128 store
  } else {
    out[i0] = accA.x;
    if (i0 + 1 < n) out[i0 + 1] = accA.y;
    if (i0 + 2 < n) out[i0 + 2] = accB.x;
  }
}

// Prep kernel: build the duplicated coefficient/weight table in global ws.
// IEEE fp32 division here matches the reference's coefficient rounding.
__global__ void build_wtab(const float *__restrict__ w,
                           v4f *__restrict__ wtab, int deg) {
  const int k = blockIdx.x * blockDim.x + threadIdx.x;
  if (k < deg) {
    const float kk = (float)k;
    const float ak = (2.0f * kk + 1.0f) / (kk + 1.0f);
    const float nbk = -kk / (kk + 1.0f);
    const float wk = w[k];
    wtab[2 * k + 0] = (v4f){wk, wk, ak, ak};
    wtab[2 * k + 1] = (v4f){nbk, nbk, 0.0f, 0.0f};
  }
}

// Runtime-degree fallback (also used if scratch is unavailable).
__global__ __launch_bounds__(256) void legendre_dyn(
    const float *__restrict__ pos, const float *__restrict__ w,
    float *__restrict__ out, int n, int deg) {
  const int i = blockIdx.x * blockDim.x + threadIdx.x;
  if (i >= n) return;
  cfp4 wc = (cfp4)(uintptr_t)w;
  const float x = pos[3 * i];
  float Pm = 0.0f, Pn = 1.0f, acc = 0.0f;
  for (int k = 0; k < deg; ++k) {
    const float kk = (float)k;
    const float inv = 1.0f / (kk + 1.0f);
    acc = __builtin_fmaf(wc[k], Pn, acc);
    const float Pp = ((2.0f * kk + 1.0f) * x * Pn - kk * Pm) * inv;
    Pm = Pn;
    Pn = Pp;
  }
  out[i] = acc;
}

extern "C" void kernel_launch(void *const *d_in, const int *in_sizes, int n_in,
                              void *d_out, int out_size, void *d_ws,
                              size_t ws_size, hipStream_t stream) {
  const float *pos = (const float *)d_in[0];  // (N,3) float32
  const float *w = (const float *)d_in[1];    // (DEGREE+1,) float32
  float *out = (float *)d_out;                // (N,) float32

  const int n = in_sizes[0] / 3;
  const int deg = in_sizes[1] - 1;

  const int block = 256;

  if (deg == DEG128 && ws_size >= 2 * DEG128 * sizeof(v4f)) {
    v4f *wtab = (v4f *)d_ws;
    build_wtab<<<1, DEG128, 0, stream>>>(w, wtab, DEG128);
    const int threads_needed = (n + 3) / 4;  // 4 elements per thread
    const int grid = (threads_needed + block - 1) / block;
    legendre128_tab<<<grid, block, 0, stream>>>(pos, wtab, out, n);
  } else {
    const int grid = (n + block - 1) / block;
    legendre_dyn<<<grid, block, 0, stream>>>(pos, w, out, n, deg);
  }
}